// MCWAUCHLoss_11089605558410
// MI455X (gfx1250) — compile-verified
//
#include <hip/hip_runtime.h>

// ---------------------------------------------------------------------------
// MCWAUCHLoss for MI455X (gfx1250, wave32).
//
// Roofline: 2 x (65536 x 256) f32 inputs = 128 MB streamed once.
// 128 MB / 23.3 TB/s ~= 5.5 us -> HBM-bound. Strategy:
//   k1: one streaming pass, b128 non-temporal loads, register accumulators
//       (3 per-column planes + 2 global scalars), hierarchical partials.
//   k2: cross-partial column reduction done on the matrix pipe with
//       V_WMMA_F32_16X16X4_F32 (ones(16x4) x P(4x16) accumulated into C),
//       then the tiny scalar epilogue.
// ---------------------------------------------------------------------------

typedef float f4  __attribute__((ext_vector_type(4)));
typedef float v2f __attribute__((ext_vector_type(2)));
typedef float v8f __attribute__((ext_vector_type(8)));

#define BROWS 65536
#define CCOLS 256
#define NBLK  256                       // kernel-1 workgroups
#define ROWS_PER_BLK (BROWS / NBLK)     // 256 rows per block
#define GP (NBLK * 4)                   // 1024 partial rows (4 rowlanes/block)
#define PLANE (GP * CCOLS)              // floats per partial plane
#define EPSV 1e-32f

// Workspace layout (floats):
//   [0          , PLANE)   : sum(s * pos)  per (partial-row, col)
//   [PLANE      , 2*PLANE) : sum(s)        per (partial-row, col)
//   [2*PLANE    , 3*PLANE) : sum(pos)      per (partial-row, col)
//   [3*PLANE    , +NBLK)   : per-block pos_loss partial sums
//   [3*PLANE+NBLK, +NBLK)  : per-block neg_loss partial sums
// Total = 3*1024*256 + 512 floats  ~= 3.1 MB

__global__ __launch_bounds__(256)
void k1_stream_partials(const float* __restrict__ out_,
                        const float* __restrict__ lab_,
                        float* __restrict__ ws) {
  const int tid     = threadIdx.x;
  const int blk     = blockIdx.x;
  const int colq    = tid & 63;   // which float4 of the 256-wide row
  const int rowlane = tid >> 6;   // 0..3
  const f4* o4 = (const f4*)out_;
  const f4* l4 = (const f4*)lab_;

  float sp[4] = {0.f, 0.f, 0.f, 0.f};   // sum s over positives
  float ss[4] = {0.f, 0.f, 0.f, 0.f};   // sum s
  float np[4] = {0.f, 0.f, 0.f, 0.f};   // count positives
  float plsum = 0.f, nlsum = 0.f;       // global BCE partials

  int r = blk * ROWS_PER_BLK + rowlane;
  for (int i = 0; i < ROWS_PER_BLK / 4; ++i, r += 4) {
    // Streaming data, used exactly once: NT hint so we don't thrash caches.
    f4 o = __builtin_nontemporal_load(&o4[r * 64 + colq]);
    f4 l = __builtin_nontemporal_load(&l4[r * 64 + colq]);
#pragma unroll
    for (int j = 0; j < 4; ++j) {
      float x = o[j];
      float y = l[j];                       // exactly 0.0f or 1.0f
      float s = 1.f / (1.f + __expf(-x));   // v_exp_f32 + rcp (trans pipe)
      plsum += y * __logf(s + EPSV);
      nlsum += (1.f - y) * __logf(1.f - s + EPSV);
      sp[j] += y * s;
      ss[j] += s;
      np[j] += y;
    }
  }

  // Per-column partials: no cross-lane traffic needed (thread owns its cols).
  float* spp = ws;
  float* ssp = ws + PLANE;
  float* npp = ws + 2 * PLANE;
  const int gp = blk * 4 + rowlane;           // partial-row index in [0, GP)
  f4 vsp = {sp[0], sp[1], sp[2], sp[3]};
  f4 vss = {ss[0], ss[1], ss[2], ss[3]};
  f4 vnp = {np[0], np[1], np[2], np[3]};
  ((f4*)spp)[gp * 64 + colq] = vsp;
  ((f4*)ssp)[gp * 64 + colq] = vss;
  ((f4*)npp)[gp * 64 + colq] = vnp;

  // Block-level LDS tree reduction for the two global scalars (deterministic).
  __shared__ float rpl[256], rnl[256];
  rpl[tid] = plsum;
  rnl[tid] = nlsum;
  __syncthreads();
  for (int s = 128; s > 0; s >>= 1) {
    if (tid < s) { rpl[tid] += rpl[tid + s]; rnl[tid] += rnl[tid + s]; }
    __syncthreads();
  }
  if (tid == 0) {
    float* base3 = ws + 3 * PLANE;
    base3[blk]        = rpl[0];
    base3[NBLK + blk] = rnl[0];
  }
}

// Kernel 2: reduce the [GP][CCOLS] planes over GP rows with the matrix unit.
// D = ones(16x4) x P(4x16) + C sums 4 partial rows of a 16-column tile per
// v_wmma_f32_16x16x4_f32; the accumulator carries the running column sums.
// A is all-ones so the exact K<->(lane-half,vgpr) permutation of the B layout
// cannot change the result. Row M=0 of D lives in VGPR0 lanes 0-15.
__global__ __launch_bounds__(256)
void k2_wmma_finalize(const float* __restrict__ ws, float* __restrict__ out) {
  const int tid  = threadIdx.x;
  const int lane = tid & 31;    // wave32
  const int wave = tid >> 5;    // 8 waves

  __shared__ float colres[3 * CCOLS];

  // ---- Phase A: WMMA column reductions. 48 tasks = 3 planes x 16 tiles,
  //      exactly 6 per wave -> every wave fully active (EXEC all-ones). ----
  const v2f aOnes  = {1.f, 1.f};
  const int colIn  = lane & 15;
  const int rowoff = (lane >> 4) * 2;
#pragma unroll 1
  for (int j = 0; j < 6; ++j) {
    const int task = wave * 6 + j;
    const int p = task >> 4;          // plane 0..2
    const int t = task & 15;          // column tile 0..15
    const float* plane = ws + p * PLANE;
    const int col = t * 16 + colIn;
    v8f acc = {};
    for (int g = 0; g < GP; g += 4) {
      v2f bv;
      bv.x = plane[(g + rowoff)     * CCOLS + col];
      bv.y = plane[(g + rowoff + 1) * CCOLS + col];
      acc = __builtin_amdgcn_wmma_f32_16x16x4_f32(
          /*neg_a=*/false, aOnes, /*neg_b=*/false, bv,
          /*c_mod=*/(short)0, acc, /*reuse_a=*/false, /*reuse_b=*/false);
    }
    if (lane < 16) colres[p * CCOLS + t * 16 + lane] = acc[0];
  }
  __syncthreads();

  // ---- Phase B: per-column penalty + final scalar combine. ----
  const int c = tid;
  const float spv = colres[c];
  const float ssv = colres[CCOLS + c];
  const float npv = colres[2 * CCOLS + c];
  const float nnv = (float)BROWS - npv;
  const float mp  = spv / fmaxf(npv, 1.f);
  const float mn  = (ssv - spv) / fmaxf(nnv, 1.f);
  float pen;
  if (npv > 0.f && nnv > 0.f)  pen = 1.f - mp + mn;
  else if (npv == 0.f)         pen = 1.f + mn;
  else                         pen = 1.f - mp;
  if (c == CCOLS - 1) out[1] = 0.1f * pen;   // penalty_term of last category

  __shared__ float r1[256], r2[256];
  __shared__ float saved[2];
  r1[c] = pen;
  r2[c] = npv;
  __syncthreads();
  for (int s = 128; s > 0; s >>= 1) {
    if (c < s) { r1[c] += r1[c + s]; r2[c] += r2[c + s]; }
    __syncthreads();
  }
  if (c == 0) { saved[0] = r1[0]; saved[1] = r2[0]; }   // pen_sum, num_P
  __syncthreads();

  const float* base3 = ws + 3 * PLANE;
  r1[c] = base3[c];          // NBLK == 256 per-block PL partials
  r2[c] = base3[NBLK + c];   // NL partials
  __syncthreads();
  for (int s = 128; s > 0; s >>= 1) {
    if (c < s) { r1[c] += r1[c + s]; r2[c] += r2[c + s]; }
    __syncthreads();
  }
  if (c == 0) {
    const float PL = r1[0], NL = r2[0];
    const float penSum = saved[0], numP = saved[1];
    const float total  = (float)BROWS * (float)CCOLS;   // exact in f32
    const float alphaP = numP / total;
    const float alphaN = 1.f - alphaP;
    const float cel = -alphaN * (PL / total) - alphaP * (NL / total);
    out[0] = cel + 0.1f * (penSum / (float)CCOLS);
  }
}

extern "C" void kernel_launch(void* const* d_in, const int* in_sizes, int n_in,
                              void* d_out, int out_size, void* d_ws, size_t ws_size,
                              hipStream_t stream) {
  const float* output = (const float*)d_in[0];
  const float* labels = (const float*)d_in[1];
  float* out = (float*)d_out;
  float* ws  = (float*)d_ws;
  (void)in_sizes; (void)n_in; (void)out_size; (void)ws_size;

  k1_stream_partials<<<NBLK, 256, 0, stream>>>(output, labels, ws);
  k2_wmma_finalize<<<1, 256, 0, stream>>>(ws, out);
}